// TravelingWave_49323404427963
// MI455X (gfx1250) — compile-verified
//
#include <hip/hip_runtime.h>
#include <math.h>

typedef __attribute__((ext_vector_type(2))) float v2f;
typedef __attribute__((ext_vector_type(8))) float v8f;

#define NSTAGES 25
#define PAD 48              // padded feature dimension (max live size 40 -> 48)
#define LDSTRIDE 52         // LDS row stride in floats (keeps b64/b128 alignment, staggers banks)
#define MAT_ELEMS (PAD*PAD)

struct GliaParams {
  const float* spW[10];  // spread W  [s,3]
  const float* spB[10];  // spread B  [s,3]
  const float* s1w[10];  // slide1 w  [3]
  const float* s1b[10];  // slide1 b  [1]
  const float* gaW[15];  // gather W  [s-2,3]
  const float* gab[15];  // gather b  [s-2]
  const float* s2w[15];  // slide2 w  [3]
  const float* s2b[15];  // slide2 b  [1]
};

__device__ __forceinline__ void stage_dims(int st, int& n_in, int& n_out) {
  if (st < 10) { n_in = 20 + 2 * st; n_out = n_in + 2; }        // spread+slide: grow
  else         { n_in = 40 - 2 * (st - 10); n_out = n_in - 2; } // gather+slide: shrink
}

// Compose (spread|gather) with slide into one affine map per stage:
//   y = x @ M + bias,   M stored TRANSPOSED: Mt[f*48 + k] = M[k][f], zero padded.
__global__ void glia_prep_kernel(GliaParams p, float* __restrict__ Mt,
                                 float* __restrict__ Bias) {
  const int st = blockIdx.x;
  int n_in, n_out; stage_dims(st, n_in, n_out);
  const int mid = n_out;                 // size the slide operates on
  const bool isSpread = (st < 10);
  const float* WA; const float* BA; const float* sw; float bb;
  if (isSpread) { WA = p.spW[st]; BA = p.spB[st]; sw = p.s1w[st]; bb = p.s1b[st][0]; }
  else { int t = st - 10; WA = p.gaW[t]; BA = p.gab[t]; sw = p.s2w[t]; bb = p.s2b[t][0]; }
  const float w[3] = {sw[0], sw[1], sw[2]};

  for (int idx = threadIdx.x; idx < MAT_ELEMS; idx += blockDim.x) {
    const int f = idx / PAD;   // output feature
    const int k = idx % PAD;   // input feature
    float acc = 0.f;
    if (f < n_out && k < n_in) {
      #pragma unroll
      for (int m = 0; m < 3; ++m) {
        const int q = f - 1 + m;           // slide tap position (zero-padded conv)
        if (q >= 0 && q < mid) {
          float a;
          if (isSpread) { const int d = q - k; a = (d >= 0 && d <= 2) ? WA[k*3 + d] : 0.f; }
          else          { const int d = k - q; a = (d >= 0 && d <= 2) ? WA[q*3 + d] : 0.f; }
          acc += a * w[m];
        }
      }
    }
    Mt[st*MAT_ELEMS + f*PAD + k] = acc;
  }
  if (threadIdx.x < PAD) {
    const int f = threadIdx.x;
    float acc = 0.f;
    if (f < n_out) {
      acc = bb;
      #pragma unroll
      for (int m = 0; m < 3; ++m) {
        const int q = f - 1 + m;
        if (q >= 0 && q < mid) {
          float c;
          if (isSpread) {                    // pre-slide bias c1[q] = sum_k B[q-k][k]
            c = 0.f;
            #pragma unroll
            for (int k2 = 0; k2 < 3; ++k2) {
              const int i = q - k2;
              if (i >= 0 && i < n_in) c += BA[i*3 + k2];
            }
          } else c = BA[q];                  // gather bias
          acc += c * w[m];
        }
      }
    }
    Bias[st*PAD + f] = acc;
  }
}

// One wave <-> one 16-row batch tile. D[M=feature][N=row] = (Mt-tile) x (X^T) via
// V_WMMA_F32_16X16X4_F32 (exact f32). Softmax = per-lane VALU + one shfl_xor(16).
__global__ void __launch_bounds__(256) glia_wave_kernel(
    const float* __restrict__ x, float* __restrict__ out,
    const float* __restrict__ Mt, const float* __restrict__ Bias, int rows) {
  __shared__ float tile[8][16 * LDSTRIDE];
  const int lane = threadIdx.x & 31;
  const int wave = threadIdx.x >> 5;
  const int n    = lane & 15;    // batch row within tile / A-row (feature)
  const int h    = lane >> 4;    // half-wave: selects K pair and feature+8
  const int rowBase = (blockIdx.x * 8 + wave) * 16;
  float* slab = tile[wave];
  (void)rows;

  // Load 16x20 input tile, zero-pad features 20..47.
  for (int idx = lane; idx < 16 * PAD; idx += 32) {
    const int r = idx / PAD, c = idx % PAD;
    slab[r * LDSTRIDE + c] = (c < 20) ? x[(size_t)(rowBase + r) * 20 + c] : 0.f;
  }
  __syncthreads();

  for (int st = 0; st < NSTAGES; ++st) {
    int n_in, n_out; stage_dims(st, n_in, n_out);
    const int kch = (n_in + 3) >> 2;     // K chunks of 4
    const int jt  = (n_out + 15) >> 4;   // 16-feature output tiles (1..3)
    const float* Mst = Mt + st * MAT_ELEMS;
    const float* Bst = Bias + st * PAD;

    auto tilemm = [&](int j) -> v8f {
      v8f c = {};
      const float* arow = Mst + (16 * j + n) * PAD + 2 * h;  // Mt[f][k..k+1]
      const float* brow = slab + n * LDSTRIDE + 2 * h;       // X [row n][k..k+1]
      for (int kc = 0; kc < kch; ++kc) {
        v2f av = *(const v2f*)(arow + 4 * kc);
        v2f bv = *(const v2f*)(brow + 4 * kc);
        c = __builtin_amdgcn_wmma_f32_16x16x4_f32(false, av, false, bv,
                                                  (short)0, c, false, false);
      }
      #pragma unroll
      for (int r = 0; r < 8; ++r) c[r] += Bst[16 * j + 8 * h + r];  // bias per feature
      return c;
    };
    v8f t0 = tilemm(0);
    v8f t1 = {}; if (jt > 1) t1 = tilemm(1);   // jt is wave-uniform: EXEC stays all-1s
    v8f t2 = {}; if (jt > 2) t2 = tilemm(2);

    if (st < NSTAGES - 1) {
      // softmax over features of row n (features split across lanes L and L+16)
      float mx = -INFINITY;
      #pragma unroll
      for (int r = 0; r < 8; ++r) {
        const int f = 8 * h + r;
        mx = (f < n_out) ? fmaxf(mx, t0[r]) : mx;
        mx = (jt > 1 && 16 + f < n_out) ? fmaxf(mx, t1[r]) : mx;
        mx = (jt > 2 && 32 + f < n_out) ? fmaxf(mx, t2[r]) : mx;
      }
      mx = fmaxf(mx, __shfl_xor(mx, 16, 32));
      float s = 0.f;
      #pragma unroll
      for (int r = 0; r < 8; ++r) {
        const int f = 8 * h + r;
        float e0 = (f < n_out) ? __expf(t0[r] - mx) : 0.f;            t0[r] = e0; s += e0;
        float e1 = (jt > 1 && 16 + f < n_out) ? __expf(t1[r] - mx) : 0.f; t1[r] = e1; s += e1;
        float e2 = (jt > 2 && 32 + f < n_out) ? __expf(t2[r] - mx) : 0.f; t2[r] = e2; s += e2;
      }
      s += __shfl_xor(s, 16, 32);
      const float inv = 1.f / s;
      #pragma unroll
      for (int r = 0; r < 8; ++r) { t0[r] *= inv; t1[r] *= inv; t2[r] *= inv; }

      // write back as next stage's X (masked lanes wrote exact 0 -> re-zeroes padding)
      {
        float* dst = slab + n * LDSTRIDE + 8 * h;
        float4 lo = {t0[0], t0[1], t0[2], t0[3]}, hi = {t0[4], t0[5], t0[6], t0[7]};
        *(float4*)(dst) = lo; *(float4*)(dst + 4) = hi;
        if (jt > 1) {
          float4 l1 = {t1[0], t1[1], t1[2], t1[3]}, h1 = {t1[4], t1[5], t1[6], t1[7]};
          *(float4*)(dst + 16) = l1; *(float4*)(dst + 20) = h1;
        }
        if (jt > 2) {
          float4 l2 = {t2[0], t2[1], t2[2], t2[3]}, h2 = {t2[4], t2[5], t2[6], t2[7]};
          *(float4*)(dst + 32) = l2; *(float4*)(dst + 36) = h2;
        }
      }
      __syncthreads();
    } else {
      // final stage: n_out = 10, jt = 1 -> log_softmax and store
      float mx = -INFINITY;
      #pragma unroll
      for (int r = 0; r < 8; ++r) {
        const int f = 8 * h + r;
        mx = (f < 10) ? fmaxf(mx, t0[r]) : mx;
      }
      mx = fmaxf(mx, __shfl_xor(mx, 16, 32));
      float s = 0.f;
      #pragma unroll
      for (int r = 0; r < 8; ++r) {
        const int f = 8 * h + r;
        s += (f < 10) ? __expf(t0[r] - mx) : 0.f;
      }
      s += __shfl_xor(s, 16, 32);
      const float lse = mx + __logf(s);
      const size_t row = (size_t)(rowBase + n);
      #pragma unroll
      for (int r = 0; r < 8; ++r) {
        const int f = 8 * h + r;
        if (f < 10) out[row * 10 + f] = t0[r] - lse;
      }
    }
  }
}

extern "C" void kernel_launch(void* const* d_in, const int* in_sizes, int n_in,
                              void* d_out, int out_size, void* d_ws, size_t ws_size,
                              hipStream_t stream) {
  (void)n_in; (void)out_size; (void)ws_size;
  GliaParams p;
  int idx = 1;  // d_in[0] = x
  for (int t = 0; t < 10; ++t) { p.spW[t] = (const float*)d_in[idx++]; p.spB[t] = (const float*)d_in[idx++]; }
  for (int t = 0; t < 10; ++t) { p.s1w[t] = (const float*)d_in[idx++]; p.s1b[t] = (const float*)d_in[idx++]; }
  for (int t = 0; t < 15; ++t) { p.gaW[t] = (const float*)d_in[idx++]; p.gab[t] = (const float*)d_in[idx++]; }
  for (int t = 0; t < 15; ++t) { p.s2w[t] = (const float*)d_in[idx++]; p.s2b[t] = (const float*)d_in[idx++]; }

  const float* x = (const float*)d_in[0];
  const int rows = in_sizes[0] / 20;            // 262144
  float* Mt   = (float*)d_ws;                   // 25 * 48*48 floats
  float* Bias = Mt + NSTAGES * MAT_ELEMS;       // 25 * 48 floats (~235 KB total)

  glia_prep_kernel<<<NSTAGES, 256, 0, stream>>>(p, Mt, Bias);
  glia_wave_kernel<<<rows / 128, 256, 0, stream>>>(x, (float*)d_out, Mt, Bias, rows);
}